// chemical_embedding_25838523252762
// MI455X (gfx1250) — compile-verified
//
#include <hip/hip_runtime.h>

// out[b, i*E + j] = x[b, i] * W[i, j]   (B=8192, L=100, E=64), f32 in/out.
// Store-bandwidth-bound (~210 MB written); WMMA F32 16x16x4 does the rank-1
// outer product per i with exact f32 semantics (only K=0 populated, C=0).

typedef __attribute__((ext_vector_type(2))) float v2f;
typedef __attribute__((ext_vector_type(8))) float v8f;

#define L_DIM 100
#define E_DIM 64
#define B_DIM 8192
#define BT    16           // WMMA M tile (batch rows per wave)
#define ROWSTRIDE (L_DIM * E_DIM)   // 6400 floats per output row

__global__ __launch_bounds__(256)
void embed_outer_wmma(const float* __restrict__ x,
                      const float* __restrict__ W,
                      float* __restrict__ out)
{
    const int lane = threadIdx.x & 31;
    const int wave = blockIdx.x * (blockDim.x >> 5) + (threadIdx.x >> 5);

    // wave -> (batch tile, embedding row i)
    const int bt = wave / L_DIM;
    const int i  = wave - bt * L_DIM;

    const int half = lane >> 4;   // 0: lanes 0-15, 1: lanes 16-31
    const int l16  = lane & 15;

    // ---- A matrix (16x4 f32): K=0 holds x[b0+m, i], K=1..3 = 0 ----
    // 32-bit A layout: VGPR0 lanes0-15 = K=0 (M=lane). Lanes 16-31 zeroed.
    const float xv = x[(size_t)(bt * BT + l16) * L_DIM + i];   // in-bounds for all lanes
    v2f a;
    a.x = (half == 0) ? xv : 0.0f;
    a.y = 0.0f;

    const float* wrow = W + i * E_DIM;

    // Per-lane output base: row = bt*16 + half*8 (+r via imm offset), col = i*64 + l16 (+j0)
    float* p = out + (size_t)(bt * BT + half * 8) * ROWSTRIDE + i * E_DIM + l16;

    #pragma unroll
    for (int j0 = 0; j0 < E_DIM; j0 += 16) {
        // ---- B matrix (4x16 f32): row K=0 holds W[i, j0+n], other rows 0 ----
        const float wv = wrow[j0 + l16];
        v2f b;
        b.x = (half == 0) ? wv : 0.0f;
        b.y = 0.0f;

        v8f c = {};
        c = __builtin_amdgcn_wmma_f32_16x16x4_f32(
                /*neg_a=*/false, a, /*neg_b=*/false, b,
                /*c_mod=*/(short)0, c, /*reuse_a=*/false, /*reuse_b=*/false);

        // D layout: VGPR r -> row (base+r), col l16 (+j0). 8 coalesced b32 stores,
        // row stride folds into the 24-bit immediate offset. Non-temporal: the
        // 210 MB output stream should not thrash the 192 MB L2.
        #pragma unroll
        for (int r = 0; r < 8; ++r) {
            __builtin_nontemporal_store(c[r], p + j0 + (size_t)r * ROWSTRIDE);
        }
    }
}

extern "C" void kernel_launch(void* const* d_in, const int* in_sizes, int n_in,
                              void* d_out, int out_size, void* d_ws, size_t ws_size,
                              hipStream_t stream)
{
    const float* x = (const float*)d_in[0];   // [8192, 100]
    const float* W = (const float*)d_in[1];   // [100, 64]
    float* out     = (float*)d_out;           // [8192, 1, 6400]

    const int total_waves = (B_DIM / BT) * L_DIM;   // 512 * 100 = 51200
    const int waves_per_block = 256 / 32;           // 8
    const int blocks = total_waves / waves_per_block; // 6400

    embed_outer_wmma<<<blocks, 256, 0, stream>>>(x, W, out);
}